// GAT_nn_2757369004092
// MI455X (gfx1250) — compile-verified
//
#include <hip/hip_runtime.h>
#include <cstdint>

typedef float v2f __attribute__((ext_vector_type(2)));
typedef float v8f __attribute__((ext_vector_type(8)));

#define NEG_SLOPE 0.2f

static __device__ __forceinline__ float atomicMaxF(float* addr, float val) {
  // signed/unsigned bit-pattern trick: valid for mixed signs
  if (val >= 0.0f)
    return __int_as_float(atomicMax((int*)addr, __float_as_int(val)));
  else
    return __uint_as_float(atomicMin((unsigned int*)addr, __float_as_uint(val)));
}

// ---------------------------------------------------------------------------
// C = A[M,K] x B[K,N], row-major, one 16x16 output tile per wave32 via
// V_WMMA_F32_16X16X4_F32 (full f32 precision).
// A 16x4 layout  : VGPR0 = K0(l0-15)|K2(l16-31), VGPR1 = K1|K3, M = lane%16
// B 4x16 layout  : VGPR0 = K0(l0-15)|K2(l16-31), VGPR1 = K1|K3, N = lane%16
// D 16x16 layout : VGPR g = M=g (l0-15) | M=g+8 (l16-31), N = lane%16
// ---------------------------------------------------------------------------
__global__ void gemm_wmma_f32(const float* __restrict__ A,
                              const float* __restrict__ B,
                              float* __restrict__ C,
                              int M, int K, int N) {
  const int lane = threadIdx.x & 31;
  const int wave = (blockIdx.x * blockDim.x + threadIdx.x) >> 5;
  const int tilesN = N >> 4;
  const int tilesM = M >> 4;
  if (wave >= tilesM * tilesN) return;            // wave-uniform: EXEC stays all-1
  const int tm = wave / tilesN;
  const int tn = wave % tilesN;
  const int row  = lane & 15;
  const int half = lane >> 4;

  const float* Ap = A + (size_t)(tm * 16 + row) * K + 2 * half;
  const float* Bp = B + (size_t)(2 * half) * N + tn * 16 + row;

  v8f acc = {};
  for (int k = 0; k < K; k += 4) {
    v2f a, b;
    a.x = Ap[k];                  // K = k + 2*half
    a.y = Ap[k + 1];              // K = k + 2*half + 1
    b.x = Bp[(size_t)k * N];      // K = k + 2*half
    b.y = Bp[(size_t)(k + 1) * N];
    acc = __builtin_amdgcn_wmma_f32_16x16x4_f32(
        /*neg_a=*/false, a, /*neg_b=*/false, b,
        /*c_mod=*/(short)0, acc, /*reuse_a=*/false, /*reuse_b=*/false);
  }

  float* Cp = C + (size_t)(tm * 16 + 8 * half) * N + tn * 16 + row;
#pragma unroll
  for (int g = 0; g < 8; ++g)
    Cp[(size_t)g * N] = acc[g];
}

// alpha_src[n] = h[n,:] . a_s ; alpha_dst[n] = h[n,:] . a_d  (one wave per node)
__global__ void alpha_dots(const float* __restrict__ h,
                           const float* __restrict__ avs,
                           const float* __restrict__ avd,
                           float* __restrict__ as_, float* __restrict__ ad_,
                           int n, int F) {
  const int lane = threadIdx.x & 31;
  const int wave = (blockIdx.x * blockDim.x + threadIdx.x) >> 5;
  if (wave >= n) return;
  float s0 = 0.f, s1 = 0.f;
  const float* hp = h + (size_t)wave * F;
  for (int f = lane; f < F; f += 32) {
    float hv = hp[f];
    s0 += hv * avs[f];
    s1 += hv * avd[f];
  }
#pragma unroll
  for (int off = 16; off > 0; off >>= 1) {
    s0 += __shfl_down(s0, off, 32);
    s1 += __shfl_down(s1, off, 32);
  }
  if (lane == 0) { as_[wave] = s0; ad_[wave] = s1; }
}

// reset per-node max/sum and zero the aggregation buffer
__global__ void init_layer(float* __restrict__ mx, float* __restrict__ sm,
                           float* __restrict__ agg, int n, long long tot) {
  long long i = (long long)blockIdx.x * blockDim.x + threadIdx.x;
  if (i < n) { mx[i] = -1e30f; sm[i] = 0.0f; }
  if (i < tot) agg[i] = 0.0f;
}

// pass 1: logits + segment max over dst
__global__ void edge_logits(const int* __restrict__ ei, int E, int n,
                            const float* __restrict__ as_,
                            const float* __restrict__ ad_,
                            float* __restrict__ ew, float* __restrict__ mx) {
  int e = blockIdx.x * blockDim.x + threadIdx.x;
  int ET = E + n;
  if (e >= ET) return;
  int s, d;
  if (e < E) { s = ei[e]; d = ei[E + e]; } else { s = d = e - E; }
  float l = as_[s] + ad_[d];
  l = (l > 0.f) ? l : NEG_SLOPE * l;
  ew[e] = l;
  atomicMaxF(&mx[d], l);
}

// pass 2: exp(logit - max) + segment sum
__global__ void edge_exp(const int* __restrict__ ei, int E, int n,
                         const float* __restrict__ mx,
                         float* __restrict__ ew, float* __restrict__ sm) {
  int e = blockIdx.x * blockDim.x + threadIdx.x;
  int ET = E + n;
  if (e >= ET) return;
  int d = (e < E) ? ei[E + e] : (e - E);
  float ex = expf(ew[e] - mx[d]);
  ew[e] = ex;
  atomicAdd(&sm[d], ex);
}

// pass 3: agg[dst,:] += h[src,:] * (w / (sum[dst]+eps))  (one wave per edge)
__global__ void edge_scatter(const int* __restrict__ ei, int E, int n, int F,
                             const float* __restrict__ h,
                             const float* __restrict__ ew,
                             const float* __restrict__ sm,
                             float* __restrict__ agg) {
  const int gid  = blockIdx.x * blockDim.x + threadIdx.x;
  const int wave = gid >> 5;
  const int lane = gid & 31;
  const int ET = E + n;
  if (wave >= ET) return;
  int s, d;
  if (wave < E) { s = ei[wave]; d = ei[E + wave]; } else { s = d = wave - E; }
  const float coef = ew[wave] / (sm[d] + 1e-16f);
  const float* hp = h + (size_t)s * F;
  float* op = agg + (size_t)d * F;
  for (int f = lane; f < F; f += 32)
    atomicAdd(&op[f], hp[f] * coef);
}

// out = agg + bias (optionally ReLU)
__global__ void bias_act(const float* __restrict__ agg,
                         const float* __restrict__ b,
                         float* __restrict__ out, int n, int F, int relu) {
  long long i = (long long)blockIdx.x * blockDim.x + threadIdx.x;
  if (i >= (long long)n * F) return;
  float v = agg[i] + b[(int)(i % F)];
  if (relu) v = fmaxf(v, 0.0f);
  out[i] = v;
}

static inline int cdiv(long long a, long long b) { return (int)((a + b - 1) / b); }

extern "C" void kernel_launch(void* const* d_in, const int* in_sizes, int n_in,
                              void* d_out, int out_size, void* d_ws, size_t ws_size,
                              hipStream_t stream) {
  const float* x   = (const float*)d_in[0];
  const int*   ei  = (const int*)d_in[1];     // [2, E] (src row then dst row)
  const float* W1  = (const float*)d_in[2];
  const float* aS1 = (const float*)d_in[3];
  const float* aD1 = (const float*)d_in[4];
  const float* b1  = (const float*)d_in[5];
  const float* W2  = (const float*)d_in[6];
  const float* aS2 = (const float*)d_in[7];
  const float* aD2 = (const float*)d_in[8];
  const float* b2  = (const float*)d_in[9];
  float* out = (float*)d_out;

  const int HID = in_sizes[3];                 // 128
  const int OUT = in_sizes[7];                 // 64
  const int IN  = in_sizes[2] / HID;           // 128
  const int N   = in_sizes[0] / IN;            // 50000
  const int E   = in_sizes[1] / 2;             // 1,600,000
  const int ET  = E + N;                       // + self-loops

  // workspace layout (floats), buffers reused across layers
  float* bufA = (float*)d_ws;                  // N*HID : h1 -> act1 -> agg2
  float* bufB = bufA + (size_t)N * HID;        // N*HID : agg1 -> h2
  float* as_  = bufB + (size_t)N * HID;
  float* ad_  = as_ + N;
  float* mx   = ad_ + N;
  float* sm   = mx + N;
  float* ew   = sm + N;                        // ET floats

  const int TB = 256;

  // ----------------------- layer 1 (F = HID) -----------------------
  {
    const int F = HID;
    int waves = (N / 16) * (F / 16);
    gemm_wmma_f32<<<cdiv((long long)waves * 32, TB), TB, 0, stream>>>(x, W1, bufA, N, IN, F);
    alpha_dots<<<cdiv((long long)N * 32, TB), TB, 0, stream>>>(bufA, aS1, aD1, as_, ad_, N, F);
    init_layer<<<cdiv((long long)N * F, TB), TB, 0, stream>>>(mx, sm, bufB, N, (long long)N * F);
    edge_logits<<<cdiv(ET, TB), TB, 0, stream>>>(ei, E, N, as_, ad_, ew, mx);
    edge_exp<<<cdiv(ET, TB), TB, 0, stream>>>(ei, E, N, mx, ew, sm);
    edge_scatter<<<cdiv((long long)ET * 32, TB), TB, 0, stream>>>(ei, E, N, F, bufA, ew, sm, bufB);
    bias_act<<<cdiv((long long)N * F, TB), TB, 0, stream>>>(bufB, b1, bufA, N, F, 1);
  }

  // ----------------------- layer 2 (F = OUT) -----------------------
  {
    const int F = OUT;
    int waves = (N / 16) * (F / 16);
    gemm_wmma_f32<<<cdiv((long long)waves * 32, TB), TB, 0, stream>>>(bufA, W2, bufB, N, HID, F);
    alpha_dots<<<cdiv((long long)N * 32, TB), TB, 0, stream>>>(bufB, aS2, aD2, as_, ad_, N, F);
    init_layer<<<cdiv((long long)N * F, TB), TB, 0, stream>>>(mx, sm, bufA, N, (long long)N * F);
    edge_logits<<<cdiv(ET, TB), TB, 0, stream>>>(ei, E, N, as_, ad_, ew, mx);
    edge_exp<<<cdiv(ET, TB), TB, 0, stream>>>(ei, E, N, mx, ew, sm);
    edge_scatter<<<cdiv((long long)ET * 32, TB), TB, 0, stream>>>(ei, E, N, F, bufB, ew, sm, bufA);
    bias_act<<<cdiv((long long)N * F, TB), TB, 0, stream>>>(bufA, b2, out, N, F, 0);
  }
}